// ReadoutFirstSpikeLayer_8246337208362
// MI455X (gfx1250) — compile-verified
//
#include <hip/hip_runtime.h>

// ReadoutFirstSpikeLayer: out[b,f] = max_t ( x[b,t,f]==1 ? T-1-t : 0 )
// x: [B=64, T=2048, F=1024] f32 binary spikes; out: [B,F] f32.
//
// Strategy (MI455X / gfx1250):
//  - Pure bandwidth-bound reduction (512 MiB in @ 23.3 TB/s ~ 23 us floor).
//  - Stream x in 16x256 tiles via the Tensor Data Mover (tensor_load_to_lds),
//    double-buffered in LDS, tracked with TENSORcnt.
//  - Forward scan => first hit is the max; block-wide early-exit vote
//    (expected ~1 tile read for Bernoulli(1/2) spikes).

#define B_DIM 64
#define T_DIM 2048
#define F_DIM 1024
#define FT    256            // features per block (= blockDim.x)
#define TS    16             // timesteps per tile
#define NTILES (T_DIM / TS)  // 128

typedef __attribute__((ext_vector_type(4))) unsigned int v4u;
typedef __attribute__((ext_vector_type(8))) int         v8i;
typedef __attribute__((ext_vector_type(4))) int         v4i;

__device__ __forceinline__ void tdm_issue(unsigned long long gaddr,
                                          unsigned lds_byte_off,
                                          const v8i& g1, const v4i& g2,
                                          const v4i& g3, const v8i& g4) {
  v4u g0;
  g0[0] = 1u;                                           // count=1, user mode, no gather
  g0[1] = lds_byte_off;                                 // LDS dest byte address
  g0[2] = (unsigned)(gaddr & 0xFFFFFFFFu);              // global_addr[31:0]
  g0[3] = (unsigned)((gaddr >> 32) & 0x01FFFFFFu)       // global_addr[56:32]
        | (2u << 30);                                   // type = 2 ("image")
  // clang-23 / therock-10.0 6-arg form:
  // (g0, g1, g2, g3, extra256, cachepolicy)
  __builtin_amdgcn_tensor_load_to_lds(g0, g1, g2, g3, g4, 0);
}

__global__ __launch_bounds__(FT)
void first_spike_kernel(const float* __restrict__ x, float* __restrict__ out) {
  __shared__ float lds_x[2][TS * FT];   // 2 x 16 KiB double buffer

  const int f0  = blockIdx.x * FT;
  const int b   = blockIdx.y;
  const int tid = threadIdx.x;

  // Byte address of x[b, 0, f0] — tile origin for t=0.
  const unsigned long long base =
      (unsigned long long)(uintptr_t)x +
      ((unsigned long long)b * T_DIM * F_DIM + (unsigned long long)f0) * 4ull;

  // Static descriptor group 1: 2D tensor, 4-byte elements,
  // tensor_dim0 = tile_dim0 = FT, tensor_dim1 = tile_dim1 = TS,
  // tensor_dim0_stride = F (row-to-row = one timestep).
  v8i g1;
  g1[0] = 0x00020000;                         // data_size=2 (4B); mask/flags = 0
  g1[1] = (FT & 0xFFFF) << 16;                // tensor_dim0[15:0] in bits[63:48]
  g1[2] = (FT >> 16) | ((TS & 0xFFFF) << 16); // tensor_dim0[31:16] | tensor_dim1[15:0]
  g1[3] = (TS >> 16) | (FT << 16);            // tensor_dim1[31:16] | tile_dim0
  g1[4] = TS;                                 // tile_dim1 (tile_dim2 = 0)
  g1[5] = F_DIM;                              // tensor_dim0_stride[31:0]
  g1[6] = 0;                                  // stride hi | dim1_stride lo (unused, 2D)
  g1[7] = 0;
  v4i g2 = {0, 0, 0, 0};                      // 2D: groups 2/3 unused
  v4i g3 = {0, 0, 0, 0};
  v8i g4 = {0, 0, 0, 0, 0, 0, 0, 0};          // extra operand of 6-arg builtin: zero

  const bool issuer = (tid < 32);             // wave 0 drives the TDM
  float res = 0.0f;

  // Prologue: tile 0 -> buffer 0.
  if (issuer) tdm_issue(base, 0u, g1, g2, g3, g4);

  for (int k = 0; k < NTILES; ++k) {
    if (issuer) __builtin_amdgcn_s_wait_tensorcnt(0);   // tile k landed in LDS
    __syncthreads();                                    // publish to all waves

    const int cur = k & 1;
    if (issuer && (k + 1) < NTILES) {                   // prefetch tile k+1
      unsigned long long ga =
          base + (unsigned long long)(k + 1) * TS * F_DIM * 4ull;
      tdm_issue(ga, (unsigned)((cur ^ 1) * (TS * FT * 4)), g1, g2, g3, g4);
    }

    if (res == 0.0f) {
      const float* buf = &lds_x[cur][0];
      const int tbase = T_DIM - 1 - k * TS;             // reverse-coded time at tt=0
      #pragma unroll
      for (int tt = 0; tt < TS; ++tt) {
        float v = buf[tt * FT + tid];
        if (res == 0.0f && v == 1.0f) res = (float)(tbase - tt);
      }
    }

    // Vote + barrier: (a) uniform early exit once every lane found its first
    // spike; (b) fences all readers of buf[cur] before it is reissued as the
    // destination of tile k+2 next iteration.
    if (__syncthreads_count(res == 0.0f) == 0) break;
  }

  out[(size_t)b * F_DIM + f0 + tid] = res;
}

extern "C" void kernel_launch(void* const* d_in, const int* in_sizes, int n_in,
                              void* d_out, int out_size, void* d_ws, size_t ws_size,
                              hipStream_t stream) {
  (void)in_sizes; (void)n_in; (void)out_size; (void)d_ws; (void)ws_size;
  const float* x = (const float*)d_in[0];
  float* out = (float*)d_out;
  dim3 grid(F_DIM / FT, B_DIM, 1);
  first_spike_kernel<<<grid, dim3(FT, 1, 1), 0, stream>>>(x, out);
}